// ConvEncoder_53480932769968
// MI455X (gfx1250) — compile-verified
//
#include <hip/hip_runtime.h>
#include <stdint.h>

typedef __attribute__((ext_vector_type(2))) float v2f;
typedef __attribute__((ext_vector_type(8))) float v8f;

#define B_ 8
#define L_ 2048
#define D_ 128
#define S_ 32
#define NTOT (B_ * L_ * D_) /* 2,097,152 */
#define RED_BLOCKS 512
#define RED_THREADS 256
#define TL 32 /* L-rows per workgroup in main kernel */

// Workspace layout (floats):
//   [0, RED_BLOCKS*9)  : per-block partial sums (Sl,Sc,Sr,Qll,Qcc,Qrr,Qlc,Qlr,Qcr)
//   [8192, 8192+128)   : fused conv+BN coefficients W'[4][32]

// ---------------------------------------------------------------------------
// Pass 1: 9 channel-independent reductions over x (with zero-padded L shifts)
// ---------------------------------------------------------------------------
__global__ __launch_bounds__(RED_THREADS) void snn_reduce_kernel(
    const float* __restrict__ x, float* __restrict__ ws) {
  __shared__ float sred[RED_THREADS * 9];
  float acc[9];
#pragma unroll
  for (int k = 0; k < 9; ++k) acc[k] = 0.f;
  const int tid = threadIdx.x;
  for (int e = blockIdx.x * RED_THREADS + tid; e < NTOT;
       e += RED_BLOCKS * RED_THREADS) {
    const int l = (e / D_) % L_;
    const float c = x[e];
    const float a = (l > 0) ? x[e - D_] : 0.f;       // x[l-1] (pad 0)
    const float r = (l < L_ - 1) ? x[e + D_] : 0.f;  // x[l+1] (pad 0)
    acc[0] += a;     acc[1] += c;     acc[2] += r;
    acc[3] += a * a; acc[4] += c * c; acc[5] += r * r;
    acc[6] += a * c; acc[7] += a * r; acc[8] += c * r;
  }
#pragma unroll
  for (int k = 0; k < 9; ++k) sred[tid * 9 + k] = acc[k];
  __syncthreads();
  for (int st = RED_THREADS / 2; st > 0; st >>= 1) {
    if (tid < st) {
#pragma unroll
      for (int k = 0; k < 9; ++k) sred[tid * 9 + k] += sred[(tid + st) * 9 + k];
    }
    __syncthreads();
  }
  if (tid < 9) ws[blockIdx.x * 9 + tid] = sred[tid];
}

// ---------------------------------------------------------------------------
// Pass 2: fold conv weights + bias + BatchNorm into affine W'[4][32]
//         y_norm(s) = W'[0][s]*x[l-1] + W'[1][s]*x[l] + W'[2][s]*x[l+1] + W'[3][s]
// ---------------------------------------------------------------------------
__global__ void snn_coef_kernel(const float* __restrict__ conv_w,
                                const float* __restrict__ conv_b,
                                const float* __restrict__ bn_gamma,
                                const float* __restrict__ bn_beta,
                                float* __restrict__ ws) {
  __shared__ float s[9];
  const int tid = threadIdx.x;
  if (tid < 9) {  // deterministic fixed-order final reduction
    float t = 0.f;
    for (int b = 0; b < RED_BLOCKS; ++b) t += ws[b * 9 + tid];
    s[tid] = t;
  }
  __syncthreads();
  if (tid < S_) {
    const float invN = 1.0f / (float)NTOT;
    const float w0 = conv_w[tid * 3 + 0];
    const float w1 = conv_w[tid * 3 + 1];
    const float w2 = conv_w[tid * 3 + 2];
    const float bb = conv_b[tid];
    const float Sl = s[0], Sc = s[1], Sr = s[2];
    const float Qll = s[3], Qcc = s[4], Qrr = s[5];
    const float Qlc = s[6], Qlr = s[7], Qcr = s[8];
    const float lin = w0 * Sl + w1 * Sc + w2 * Sr;
    const float mean = lin * invN + bb;
    const float ey2 =
        (w0 * w0 * Qll + w1 * w1 * Qcc + w2 * w2 * Qrr +
         2.f * (w0 * w1 * Qlc + w0 * w2 * Qlr + w1 * w2 * Qcr) +
         2.f * bb * lin) * invN + bb * bb;
    const float var = ey2 - mean * mean;  // biased variance, batch stats
    const float g = bn_gamma[tid] * rsqrtf(var + 1e-5f);
    float* wc = ws + 8192;
    wc[0 * S_ + tid] = g * w0;
    wc[1 * S_ + tid] = g * w1;
    wc[2 * S_ + tid] = g * w2;
    wc[3 * S_ + tid] = g * (bb - mean) + bn_beta[tid];
  }
}

// ---------------------------------------------------------------------------
// Pass 3: streaming conv+BN via V_WMMA_F32_16X16X4_F32, fused 32-step LIF,
//         coalesced spike stores. One workgroup = one (b, 32-L-row) tile.
// x tile staged with gfx1250 GLOBAL_LOAD_ASYNC_TO_LDS_B128 (ASYNCcnt path).
// ---------------------------------------------------------------------------
__global__ __launch_bounds__(256) void snn_main_kernel(
    const float* __restrict__ x, const float* __restrict__ ws,
    const float* __restrict__ thresh_p, float* __restrict__ out) {
  __shared__ float xs[(TL + 2) * D_];   // x tile incl. l-halo   (17408 B)
  __shared__ float ytr[8][32 * 35];     // per-wave transpose    (35840 B)

  const int tid = threadIdx.x;
  const int wg = blockIdx.x;
  const int b = wg >> 6;            // 64 L-tiles per batch
  const int l0 = (wg & 63) * TL;

  // Stage x[l0-1 .. l0+TL][0..127] into LDS. In-range rows use the CDNA5
  // async global->LDS DMA path (no VGPR data movement, tracked by ASYNCcnt);
  // out-of-range halo rows (L boundary) are zero-filled with plain ds stores.
  // Note: flat->LDS aperture mapping keeps the address's low 32 bits, so the low
  // 32 bits of a generic __shared__ pointer are the LDS byte offset.
  const float* xb = x + (long)b * (L_ * D_);
  {
    const int nvec = ((TL + 2) * D_) / 4;  // 1088 x b128 transfers
    for (int i4 = tid; i4 < nvec; i4 += 256) {
      const int row = i4 >> 5;         // 32 vec4 per 128-float row
      const int col = (i4 & 31) << 2;  // float column within row
      const int gl = l0 - 1 + row;
      float* lp = &xs[i4 << 2];
      if (gl >= 0 && gl < L_) {
        const unsigned ldsOff = (unsigned)(uintptr_t)lp;
        const float* gp = xb + (long)gl * D_ + col;
        asm volatile("global_load_async_to_lds_b128 %0, %1, off"
                     :: "v"(ldsOff), "v"(gp)
                     : "memory");
      } else {
        lp[0] = 0.f; lp[1] = 0.f; lp[2] = 0.f; lp[3] = 0.f;
      }
    }
    asm volatile("s_wait_asynccnt 0x0" ::: "memory");
  }

  const float thr = thresh_p[0];
  const int lane = tid & 31;
  const int wave = tid >> 5;
  const int nlo = lane & 15;

  // B matrices (4x16 f32). Layout mirrors the documented A 16x4 layout
  // transposed: VGPR0 = {K0 rows in lanes 0-15, K2 rows in lanes 16-31},
  // VGPR1 = {K1, K3}. N = lane%16 (+16 channel offset for B1).
  const float* wc = ws + 8192;
  v2f B0, B1;
  if (lane < 16) {
    B0.x = wc[0 * S_ + nlo];      B0.y = wc[1 * S_ + nlo];
    B1.x = wc[0 * S_ + 16 + nlo]; B1.y = wc[1 * S_ + 16 + nlo];
  } else {
    B0.x = wc[2 * S_ + nlo];      B0.y = wc[3 * S_ + nlo];
    B1.x = wc[2 * S_ + 16 + nlo]; B1.y = wc[3 * S_ + 16 + nlo];
  }
  __syncthreads();

  float* yt = ytr[wave];
  const long sStride = (long)B_ * L_ * D_;  // spike channel-plane stride

  // Each wave owns 512 locations = rows [4*wave, 4*wave+4) of the tile.
  for (int ph = 0; ph < 16; ++ph) {
    const int p0 = wave * 512 + ph * 32;  // 32 locations per phase

#pragma unroll
    for (int sub = 0; sub < 2; ++sub) {   // two 16-location M-tiles
      const int pm = p0 + sub * 16;
      const int p = pm + nlo;             // location index in tile (l*128+d)
      // A 16x4: lanes 0-15 hold {K0=x[l-1], K1=x[l]}, lanes 16-31 {K2=x[l+1], K3=1}
      v2f A;
      if (lane < 16) { A.x = xs[p];       A.y = xs[p + 128]; }
      else           { A.x = xs[p + 256]; A.y = 1.0f;        }
      v8f zero = {};
      v8f c0 = __builtin_amdgcn_wmma_f32_16x16x4_f32(
          false, A, false, B0, (short)0, zero, false, false);
      v8f c1 = __builtin_amdgcn_wmma_f32_16x16x4_f32(
          false, A, false, B1, (short)0, zero, false, false);
      // D layout: lane n holds channel n, rows r (lanes 16-31: rows 8+r).
      // Transpose into [location][channel] scratch (stride 35: conflict-free).
      const int locb = sub * 16 + (lane >> 4) * 8;
#pragma unroll
      for (int r = 0; r < 8; ++r) {
        yt[(locb + r) * 35 + nlo]      = c0[r];
        yt[(locb + r) * 35 + 16 + nlo] = c1[r];
      }
    }
    __builtin_amdgcn_wave_barrier();  // in-wave LDS ops are HW-ordered

    // LIF: lane == location; 32 sequential channel steps, coalesced stores.
    const int p = p0 + lane;
    const int l = l0 + (p >> 7);
    const int d = p & 127;
    float* op = out + ((long)b * L_ + l) * D_ + d;
    float v = 0.f;
#pragma unroll
    for (int s = 0; s < S_; ++s) {
      v = 0.5f * (v + yt[lane * 35 + s]);   // v += (x - v)/tau, tau = 2
      const float sp = (v >= thr) ? 1.0f : 0.0f;
      op[(long)s * sStride] = sp;
      v = (v >= thr) ? 0.0f : v;            // hard reset to 0
    }
    __builtin_amdgcn_wave_barrier();
  }
}

// ---------------------------------------------------------------------------
extern "C" void kernel_launch(void* const* d_in, const int* in_sizes, int n_in,
                              void* d_out, int out_size, void* d_ws,
                              size_t ws_size, hipStream_t stream) {
  const float* x        = (const float*)d_in[0];
  const float* conv_w   = (const float*)d_in[1];
  const float* conv_b   = (const float*)d_in[2];
  const float* bn_gamma = (const float*)d_in[3];
  const float* bn_beta  = (const float*)d_in[4];
  const float* thresh   = (const float*)d_in[5];
  float* ws  = (float*)d_ws;
  float* out = (float*)d_out;

  hipLaunchKernelGGL(snn_reduce_kernel, dim3(RED_BLOCKS), dim3(RED_THREADS), 0,
                     stream, x, ws);
  hipLaunchKernelGGL(snn_coef_kernel, dim3(1), dim3(64), 0, stream, conv_w,
                     conv_b, bn_gamma, bn_beta, ws);
  hipLaunchKernelGGL(snn_main_kernel, dim3(B_ * (L_ / TL)), dim3(256), 0,
                     stream, x, ws, thresh, out);
}